// HistBranch_16939351016189
// MI455X (gfx1250) — compile-verified
//
#include <hip/hip_runtime.h>
#include <hip/hip_bf16.h>

#define NBINS 32
#define MID 64
#define NUM_ITERS 4
#define BATCH 32
#define NPIX (1024 * 1024)           // pixels per sample
#define NPIX4 (NPIX / 4)             // float4s per sample (262144)
#define CHUNK4 4096                  // float4s per block in reduction/hist passes
#define BLOCKS_PER_SAMPLE (NPIX4 / CHUNK4)  // 64

typedef __attribute__((ext_vector_type(2))) float v2f;
typedef __attribute__((ext_vector_type(8))) float v8f;

// ---- orderable-uint encoding for float atomic min/max (handles sign) ----
__device__ __forceinline__ unsigned encode_f(float f) {
    unsigned u = __float_as_uint(f);
    return (u & 0x80000000u) ? ~u : (u ^ 0x80000000u);
}
__device__ __forceinline__ float decode_key(unsigned k) {
    unsigned u = (k & 0x80000000u) ? (k ^ 0x80000000u) : ~k;
    return __uint_as_float(u);
}

// ---------------- init workspace (deterministic each launch) ----------------
__global__ void __launch_bounds__(256) k_init(unsigned* mmkeys, unsigned* counts) {
    int tid = blockIdx.x * 256 + threadIdx.x;
    if (tid < 2 * BATCH) mmkeys[tid] = (tid & 1) ? 0u : 0xFFFFFFFFu;  // minkey=UINT_MAX, maxkey=0
    if (tid < BATCH * NBINS) counts[tid] = 0u;
}

// ---------------- pass 1: per-sample min/max ----------------
__global__ void __launch_bounds__(256) k_minmax(const float4* __restrict__ V4,
                                                unsigned* __restrict__ mmkeys) {
    const int s = blockIdx.y;
    const int tid = threadIdx.x;
    const float4* base = V4 + (size_t)s * NPIX4 + (size_t)blockIdx.x * CHUNK4;
    float vmin = 3.402823466e38f, vmax = -3.402823466e38f;
    for (int i = tid; i < CHUNK4; i += 256) {
        float4 v = base[i];
        vmin = fminf(vmin, fminf(fminf(v.x, v.y), fminf(v.z, v.w)));
        vmax = fmaxf(vmax, fmaxf(fmaxf(v.x, v.y), fmaxf(v.z, v.w)));
    }
    __shared__ float smin[256];
    __shared__ float smax[256];
    smin[tid] = vmin; smax[tid] = vmax;
    __syncthreads();
    for (int st = 128; st > 0; st >>= 1) {
        if (tid < st) {
            smin[tid] = fminf(smin[tid], smin[tid + st]);
            smax[tid] = fmaxf(smax[tid], smax[tid + st]);
        }
        __syncthreads();
    }
    if (tid == 0) {
        atomicMin(&mmkeys[2 * s + 0], encode_f(smin[0]));
        atomicMax(&mmkeys[2 * s + 1], encode_f(smax[0]));
    }
}

// ---------------- pass 2: per-sample 32-bin histogram ----------------
__global__ void __launch_bounds__(256) k_hist(const float4* __restrict__ V4,
                                              const unsigned* __restrict__ mmkeys,
                                              unsigned* __restrict__ counts) {
    const int s = blockIdx.y;
    const int tid = threadIdx.x;
    const float mn = decode_key(mmkeys[2 * s + 0]);
    const float mx = decode_key(mmkeys[2 * s + 1]);
    const float rng = mx - mn;
    const float inv = (float)NBINS / (rng > 0.0f ? rng : 1.0f);

    __shared__ unsigned lhist[8 * NBINS];   // per-wave sub-histograms
    lhist[tid] = 0u;                        // 256 entries exactly
    __syncthreads();

    unsigned* myh = &lhist[(tid >> 5) * NBINS];
    const float4* base = V4 + (size_t)s * NPIX4 + (size_t)blockIdx.x * CHUNK4;
    for (int i = tid; i < CHUNK4; i += 256) {
        float4 v = base[i];
        int i0 = (int)((v.x - mn) * inv); i0 = i0 < 0 ? 0 : (i0 > NBINS - 1 ? NBINS - 1 : i0);
        int i1 = (int)((v.y - mn) * inv); i1 = i1 < 0 ? 0 : (i1 > NBINS - 1 ? NBINS - 1 : i1);
        int i2 = (int)((v.z - mn) * inv); i2 = i2 < 0 ? 0 : (i2 > NBINS - 1 ? NBINS - 1 : i2);
        int i3 = (int)((v.w - mn) * inv); i3 = i3 < 0 ? 0 : (i3 > NBINS - 1 ? NBINS - 1 : i3);
        atomicAdd(&myh[i0], 1u);
        atomicAdd(&myh[i1], 1u);
        atomicAdd(&myh[i2], 1u);
        atomicAdd(&myh[i3], 1u);
    }
    __syncthreads();
    if (tid < NBINS) {
        unsigned t = 0;
        #pragma unroll
        for (int w = 0; w < 8; ++w) t += lhist[w * NBINS + tid];
        atomicAdd(&counts[s * NBINS + tid], t);
    }
}

// ---------------- WMMA helpers (f32 16x16x4, wave32) ----------------
// Zero-padded cooperative weight stage: dst[Kpad][Npad] <- W[K][N]
__device__ __forceinline__ void load_wpad(float* dst, const float* __restrict__ W,
                                          int K, int N, int Kpad, int Npad) {
    for (int i = threadIdx.x; i < Kpad * Npad; i += 256) {
        int k = i / Npad, n = i % Npad;
        dst[i] = (k < K && n < N) ? W[k * N + n] : 0.0f;
    }
}

// One 16x16 output tile per wave; acc over Kpad in steps of 4 via v_wmma_f32_16x16x4_f32.
// A: LDS [32][lda] row-major activations. Wb: LDS [Kpad][ldb] row-major weights.
// Dst: LDS [32][ldd]; fused bias + leaky-relu(0.01).
__device__ __forceinline__ void mlp_gemm(const float* A, int lda,
                                         const float* Wb, int ldb,
                                         const float* __restrict__ bias, int biasN,
                                         float* Dst, int ldd,
                                         int Kpad, int nTiles) {
    const int tid = threadIdx.x;
    const int wave = tid >> 5;
    const int lane = tid & 31;
    if (wave < 2 * nTiles) {                 // whole-wave predicate: EXEC stays all-ones
        const int mBase = (wave / nTiles) << 4;
        const int nBase = (wave % nTiles) << 4;
        const int half = lane >> 4;          // 0: lanes 0-15, 1: lanes 16-31
        const int l15 = lane & 15;
        const int row = mBase + l15;         // A fragment row (M)
        const int col = nBase + l15;         // B/D fragment column (N)
        const int kOff = half * 2;           // A: K={0,1}|{2,3}; B symmetric
        v8f acc = {};
        for (int k0 = 0; k0 < Kpad; k0 += 4) {
            v2f a, b;
            a.x = A[row * lda + k0 + kOff];
            a.y = A[row * lda + k0 + kOff + 1];
            b.x = Wb[(k0 + kOff) * ldb + col];
            b.y = Wb[(k0 + kOff + 1) * ldb + col];
            acc = __builtin_amdgcn_wmma_f32_16x16x4_f32(false, a, false, b,
                                                        (short)0, acc, false, false);
        }
        const float bv = (l15 + nBase < biasN) ? bias[col] : 0.0f;
        const int rBase = mBase + half * 8;  // D: lanes 0-15 hold M=mBase+0..7, 16-31 M=+8..15
        #pragma unroll
        for (int r = 0; r < 8; ++r) {
            float x = acc[r] + bv;
            Dst[(rBase + r) * ldd + col] = (x > 0.0f) ? x : 0.01f * x;
        }
    }
}

// ---------------- pass 3: feature build + 5-layer MLP (single block) ----------------
__global__ void __launch_bounds__(256) k_mlp(const unsigned* __restrict__ mmkeys,
                                             const unsigned* __restrict__ counts,
                                             const float* __restrict__ mu,
                                             const float* __restrict__ W1, const float* __restrict__ b1,
                                             const float* __restrict__ W2, const float* __restrict__ b2,
                                             const float* __restrict__ W3, const float* __restrict__ b3,
                                             const float* __restrict__ W4, const float* __restrict__ b4,
                                             const float* __restrict__ W5, const float* __restrict__ b5,
                                             float* __restrict__ alphas) {
    __shared__ float vecp[BATCH * 36];      // vec [32][35] zero-padded to K=36; reused for alphas
    __shared__ float hbuf[BATCH * 64];      // hidden activations
    __shared__ float cbuf[BATCH * 100];     // concat [h(64) | vec(35) | pad] -> K=100
    __shared__ float wbuf[100 * 64];        // staged zero-padded weights (reused per layer)
    __shared__ float sums[BATCH];
    const int tid = threadIdx.x;

    if (tid < BATCH) {
        float t = 0.0f;
        for (int j = 0; j < NBINS; ++j) t += (float)counts[tid * NBINS + j];
        sums[tid] = t;
    }
    __syncthreads();

    for (int i = tid; i < BATCH * 36; i += 256) {
        int s = i / 36, c = i % 36;
        float v;
        if (c < NBINS)      v = (float)counts[s * NBINS + c] / sums[s];
        else if (c == 32)   v = decode_key(mmkeys[2 * s + 0]);   // mn
        else if (c == 33)   v = decode_key(mmkeys[2 * s + 1]);   // mx
        else if (c == 34)   v = mu[s];
        else                v = 0.0f;                            // K pad
        vecp[i] = v;
        if (c < 35) cbuf[s * 100 + 64 + c] = v;                  // concat tail
        else        cbuf[s * 100 + 99] = 0.0f;                   // concat K pad
    }

    // layer 1: [32,36] x [36,64]
    load_wpad(wbuf, W1, 35, 64, 36, 64);
    __syncthreads();
    mlp_gemm(vecp, 36, wbuf, 64, b1, 64, hbuf, 64, 36, 4);
    __syncthreads();
    // layer 2: [32,64] x [64,64] -> cbuf cols 0..63
    load_wpad(wbuf, W2, 64, 64, 64, 64);
    __syncthreads();
    mlp_gemm(hbuf, 64, wbuf, 64, b2, 64, cbuf, 100, 64, 4);
    __syncthreads();
    // layer 3: [32,100] x [100,64]
    load_wpad(wbuf, W3, 99, 64, 100, 64);
    __syncthreads();
    mlp_gemm(cbuf, 100, wbuf, 64, b3, 64, hbuf, 64, 100, 4);
    __syncthreads();
    // layer 4: [32,64] x [64,64] -> cbuf cols 0..63
    load_wpad(wbuf, W4, 64, 64, 64, 64);
    __syncthreads();
    mlp_gemm(hbuf, 64, wbuf, 64, b4, 64, cbuf, 100, 64, 4);
    __syncthreads();
    // layer 5: [32,64] x [64,16-padded(N=4)] -> vecp reused as [32][36]
    load_wpad(wbuf, W5, 64, 4, 64, 16);
    __syncthreads();
    mlp_gemm(cbuf, 100, wbuf, 16, b5, 4, vecp, 36, 64, 1);
    __syncthreads();

    if (tid < BATCH * NUM_ITERS) {
        int s = tid >> 2, i = tid & 3;
        alphas[s * NUM_ITERS + i] = vecp[s * 36 + i];
    }
}

// ---------------- pass 4: 4-iteration logistic-ish update ----------------
__device__ __forceinline__ float iter4(float x, float a0, float a1, float a2, float a3) {
    x = fmaf(a0, fmaf(-x, x, x), x);   // x + a*(x - x*x)
    x = fmaf(a1, fmaf(-x, x, x), x);
    x = fmaf(a2, fmaf(-x, x, x), x);
    x = fmaf(a3, fmaf(-x, x, x), x);
    return x;
}

__global__ void __launch_bounds__(256) k_apply(const float4* __restrict__ V4,
                                               const float* __restrict__ alphas,
                                               float4* __restrict__ out4) {
    const unsigned i4 = blockIdx.x * 256u + threadIdx.x;
    const int s = (int)(i4 >> 18);              // NPIX4 = 2^18 float4 per sample
    const float a0 = alphas[s * 4 + 0];
    const float a1 = alphas[s * 4 + 1];
    const float a2 = alphas[s * 4 + 2];
    const float a3 = alphas[s * 4 + 3];
    float4 v = V4[i4];
    v.x = iter4(v.x, a0, a1, a2, a3);
    v.y = iter4(v.y, a0, a1, a2, a3);
    v.z = iter4(v.z, a0, a1, a2, a3);
    v.w = iter4(v.w, a0, a1, a2, a3);
    out4[i4] = v;
}

extern "C" void kernel_launch(void* const* d_in, const int* in_sizes, int n_in,
                              void* d_out, int out_size, void* d_ws, size_t ws_size,
                              hipStream_t stream) {
    const float* V  = (const float*)d_in[0];
    const float* mu = (const float*)d_in[1];
    const float* W1 = (const float*)d_in[2];
    const float* b1 = (const float*)d_in[3];
    const float* W2 = (const float*)d_in[4];
    const float* b2 = (const float*)d_in[5];
    const float* W3 = (const float*)d_in[6];
    const float* b3 = (const float*)d_in[7];
    const float* W4 = (const float*)d_in[8];
    const float* b4 = (const float*)d_in[9];
    const float* W5 = (const float*)d_in[10];
    const float* b5 = (const float*)d_in[11];
    float* out = (float*)d_out;

    unsigned* mmkeys = (unsigned*)d_ws;              // 64 uints
    unsigned* counts = mmkeys + 2 * BATCH;           // 1024 uints
    float*    alphas = (float*)(counts + BATCH * NBINS);  // 128 floats

    k_init<<<4, 256, 0, stream>>>(mmkeys, counts);
    dim3 g(BLOCKS_PER_SAMPLE, BATCH);
    k_minmax<<<g, 256, 0, stream>>>((const float4*)V, mmkeys);
    k_hist<<<g, 256, 0, stream>>>((const float4*)V, mmkeys, counts);
    k_mlp<<<1, 256, 0, stream>>>(mmkeys, counts, mu,
                                 W1, b1, W2, b2, W3, b3, W4, b4, W5, b5, alphas);
    const int nblk = (BATCH * NPIX4) / 256;          // 32768
    k_apply<<<nblk, 256, 0, stream>>>((const float4*)V, alphas, (float4*)out);
}